// HilbertCurveAttention_66425964200586
// MI455X (gfx1250) — compile-verified
//
#include <hip/hip_runtime.h>
#include <math.h>

#define BATCH   128
#define SEQ_LEN 4096
#define HIDDEN  512
#define GRID_N  256
#define WINDOW  7
#define WIN2    49
#define NTHREADS 128   // 4 waves (wave32)

typedef __attribute__((ext_vector_type(2))) float v2f;
typedef __attribute__((ext_vector_type(8))) float v8f;

__global__ __launch_bounds__(NTHREADS)
void HilbertCurveAttention_kernel(
    const float* __restrict__ query,          // (B,1,H)
    const float* __restrict__ key,            // (B,S,H)
    const float* __restrict__ value,          // (B,S,H)
    const int*   __restrict__ query_pos,      // (B,)
    const float* __restrict__ distance_scale, // (1,)
    const float* __restrict__ local_bias,     // (49,)
    const int*   __restrict__ coords,         // (65536,2)
    const int*   __restrict__ inv,            // (256,256)
    float*       __restrict__ out)            // (B,H)
{
    __shared__ __align__(16) float s_q[HIDDEN];
    __shared__ int   s_pos[64];
    __shared__ float s_dist[64];
    __shared__ float s_logw[64];
    __shared__ float s_scores[64];
    __shared__ float s_attn[64];

    const int b   = blockIdx.x;
    const int tid = threadIdx.x;

    // ---- Stage query row into LDS ----
    const float* qrow = query + (size_t)b * HIDDEN;
    for (int h = tid; h < HIDDEN; h += NTHREADS) s_q[h] = qrow[h];

    // ---- Window metadata: position gather + validity + Manhattan distance ----
    if (tid < 64) {
        int pos = -1; float dist = 0.f;
        if (tid < WIN2) {
            int qp = query_pos[b];
            int qx = coords[2 * qp];
            int qy = coords[2 * qp + 1];
            int dx = tid / WINDOW - (WINDOW / 2);   // dx outer loop order
            int dy = tid % WINDOW - (WINDOW / 2);   // dy inner loop order
            int nx = qx + dx, ny = qy + dy;
            dist = (float)(abs(dx) + abs(dy));
            if (nx >= 0 && nx < GRID_N && ny >= 0 && ny < GRID_N) {
                int p = inv[nx * GRID_N + ny];
                if (p < SEQ_LEN) pos = p;
            }
        }
        s_pos[tid]  = pos;
        s_dist[tid] = dist;
    }
    __syncthreads();

    // ---- Sequential compacted-bias log-weights (cumsum order dependency) ----
    if (tid == 0) {
        float sc = distance_scale[0];
        int c = 0;
        for (int w = 0; w < WIN2; ++w) {
            if (s_pos[w] >= 0) {
                float bias = local_bias[c++];
                float wgt  = expf(-s_dist[w] * sc) + bias;
                s_logw[w]  = logf(wgt + 1e-8f);
            } else {
                s_logw[w] = 0.f;
            }
        }
    }
    __syncthreads();

    // ---- Scores via V_WMMA_F32_16X16X4_F32: 4 waves x one 16-row M-tile ----
    // A: 16x4 fp32 tile of gathered K rows; lane l holds row (l&15), K-pair 2*(l>>4).
    // B: q broadcast across all 16 N columns -> D[m][n] = score[m] for every n.
    const int wv   = tid >> 5;            // wave 0..3 -> rows [16wv, 16wv+16)
    const int lane = tid & 31;
    const int mrow = (wv << 4) + (lane & 15);
    const int koff = (lane >> 4) << 1;    // 0 for lanes 0-15, 2 for lanes 16-31
    int prow = s_pos[mrow]; if (prow < 0) prow = 0;   // dummy row, masked later
    const float* krow = key + ((size_t)b * SEQ_LEN + (size_t)prow) * HIDDEN;

    v8f acc = {0.f, 0.f, 0.f, 0.f, 0.f, 0.f, 0.f, 0.f};
#pragma unroll 4
    for (int kb = 0; kb < HIDDEN; kb += 4) {
        v2f a  = *(const v2f*)(krow + kb + koff);   // 8B global gather
        v2f bq = *(const v2f*)(s_q  + kb + koff);   // 8B LDS read
        acc = __builtin_amdgcn_wmma_f32_16x16x4_f32(
                  false, a, false, bq, (short)0, acc, false, false);
    }
    // D layout: VGPR j -> M=j (lanes 0-15) / M=j+8 (lanes 16-31); replicated over N.
    if (lane == 0) {
#pragma unroll
        for (int j = 0; j < 8; ++j) s_scores[(wv << 4) + j] = acc[j];
    } else if (lane == 16) {
#pragma unroll
        for (int j = 0; j < 8; ++j) s_scores[(wv << 4) + 8 + j] = acc[j];
    }
    __syncthreads();

    // ---- Softmax over 49 entries (serial, tiny) ----
    if (tid == 0) {
        const float rs = 0.04419417382415922f;   // 1/sqrt(512)
        float mx = -INFINITY;
        for (int w = 0; w < WIN2; ++w) {
            float sv = (s_pos[w] >= 0) ? (s_scores[w] * rs + s_logw[w]) : -INFINITY;
            s_scores[w] = sv;
            if (sv > mx) mx = sv;
        }
        float sum = 0.f;
        for (int w = 0; w < WIN2; ++w) {
            float e = (s_pos[w] >= 0) ? expf(s_scores[w] - mx) : 0.f;
            s_attn[w] = e;
            sum += e;
        }
        float is = 1.f / sum;
        for (int w = 0; w < WIN2; ++w) s_attn[w] *= is;
    }
    __syncthreads();

    // ---- Output: coalesced float4 weighted sum over value rows ----
    const float* vbase = value + (size_t)b * SEQ_LEN * HIDDEN;
    const int h4 = tid << 2;                      // 128 threads x 4 floats = 512
    float ax = 0.f, ay = 0.f, az = 0.f, aw = 0.f;
    for (int w = 0; w < WIN2; ++w) {
        int p = s_pos[w];
        if (p < 0) continue;                      // uniform branch
        float a = s_attn[w];
        const float4 v = *(const float4*)(vbase + (size_t)p * HIDDEN + h4);
        ax += a * v.x; ay += a * v.y; az += a * v.z; aw += a * v.w;
    }
    float4 r; r.x = ax; r.y = ay; r.z = az; r.w = aw;
    *(float4*)(out + (size_t)b * HIDDEN + h4) = r;
}

extern "C" void kernel_launch(void* const* d_in, const int* in_sizes, int n_in,
                              void* d_out, int out_size, void* d_ws, size_t ws_size,
                              hipStream_t stream) {
    const float* query          = (const float*)d_in[0];
    const float* key            = (const float*)d_in[1];
    const float* value          = (const float*)d_in[2];
    const int*   query_pos      = (const int*)  d_in[3];
    const float* distance_scale = (const float*)d_in[4];
    const float* local_bias     = (const float*)d_in[5];
    const int*   coords         = (const int*)  d_in[6];
    const int*   inv            = (const int*)  d_in[7];
    float*       out            = (float*)d_out;

    HilbertCurveAttention_kernel<<<BATCH, NTHREADS, 0, stream>>>(
        query, key, value, query_pos, distance_scale, local_bias, coords, inv, out);
}